// CodecModel_15238543966362
// MI455X (gfx1250) — compile-verified
//
#include <hip/hip_runtime.h>
#include <math.h>

// ---------------------------------------------------------------------------
// Reference shapes
//   F = N = 2048, K = 32, M = (F/2)*(N/2) = 1,048,576
//   out: complex64 [F, N] -> interleaved float pairs in d_out (2*F*N floats)
// ---------------------------------------------------------------------------
#define F_DIM 2048
#define N_DIM 2048
#define K_DIM 32
#define M_DIM ((F_DIM / 2) * (N_DIM / 2))
#define INV_PI 0.31830988618379067154f

typedef __attribute__((ext_vector_type(2))) float v2f;
typedef __attribute__((ext_vector_type(4))) float v4f;
typedef __attribute__((ext_vector_type(8))) float v8f;

// ---------------------------------------------------------------------------
// Zero-fill output (harness poisons d_out with 0xAA before timing)
// ---------------------------------------------------------------------------
__global__ __launch_bounds__(256) void zero_out_kernel(v4f* __restrict__ out, int n4) {
  int i = blockIdx.x * blockDim.x + threadIdx.x;
  if (i < n4) out[i] = (v4f){0.f, 0.f, 0.f, 0.f};
}

// ---------------------------------------------------------------------------
// Main splat kernel.
//   - 256 threads/block = 8 wave32; each wave owns 32 consecutive centres.
//   - Per lane: argmax over K=32 of its centre's alpha row (streamed NT).
//   - Per wave: one-hot(k) @ P_bank[K,9->16] via 2 tiles x 8 chained
//     V_WMMA_F32_16X16X4_F32 (exact in f32), D tiles transposed through LDS.
//   - Per lane: separable bilinear merge 3x3 taps -> 4x4 patch, then 32
//     agent-scope f32 atomics into the interleaved complex grid (L2-resident).
// ---------------------------------------------------------------------------
__global__ __launch_bounds__(256) void codec_splat_kernel(
    const float* __restrict__ P_bank,     // [K,3,3] = [32*9]
    const float* __restrict__ alpha,      // [M,K]
    const float* __restrict__ zeta_f,     // [M]
    const float* __restrict__ zeta_n,     // [M]
    const float* __restrict__ log_rho,    // [M]
    const float* __restrict__ theta,      // [M]
    const float* __restrict__ f_centres,  // [M]
    const float* __restrict__ n_centres,  // [M]
    float* __restrict__ out)              // [F*N*2] interleaved re,im
{
  __shared__ float smem[8 * 32 * 17];     // per-wave 32x16 tile, stride 17 (bank-conflict free)

  const int lane = threadIdx.x & 31;
  const int wave = threadIdx.x >> 5;
  const int m    = blockIdx.x * 256 + wave * 32 + lane;   // this lane's centre

  const int col  = lane & 15;             // WMMA column / row-within-tile id
  const int koff = (lane < 16) ? 0 : 2;   // ISA 16x4 f32 layout: low lanes K=0/1, high lanes K=2/3

  // ---- B matrix (P_bank reshaped [32 x 9], cols 9..15 zero) held in VGPRs ----
  v2f B[8];
#pragma unroll
  for (int c = 0; c < 8; ++c) {
    float b0 = 0.f, b1 = 0.f;
    if (col < 9) {
      b0 = P_bank[(4 * c + koff)     * 9 + col];
      b1 = P_bank[(4 * c + koff + 1) * 9 + col];
    }
    B[c][0] = b0;
    B[c][1] = b1;
  }

  // ---- argmax over K=32 (first-max wins, matching jnp.argmax) ----
  const v4f* arow = (const v4f*)(alpha + (size_t)m * K_DIM);
  float best = -INFINITY;
  int   bk   = 0;
#pragma unroll
  for (int q = 0; q < 8; ++q) {
    v4f v = __builtin_nontemporal_load(arow + q);   // stream: don't pollute L2
    if (v[0] > best) { best = v[0]; bk = 4 * q + 0; }
    if (v[1] > best) { best = v[1]; bk = 4 * q + 1; }
    if (v[2] > best) { best = v[2]; bk = 4 * q + 2; }
    if (v[3] > best) { best = v[3]; bk = 4 * q + 3; }
  }

  // ---- one-hot @ P_bank via WMMA f32 16x16x4, two 16-centre tiles ----
  float* sm = smem + wave * (32 * 17);
#pragma unroll
  for (int t = 0; t < 2; ++t) {
    const int ks = __shfl(bk, t * 16 + col, 32);       // k of the row this lane carries
    v8f acc = {};
#pragma unroll
    for (int c = 0; c < 8; ++c) {
      v2f a;
      a[0] = (ks == 4 * c + koff)     ? 1.f : 0.f;
      a[1] = (ks == 4 * c + koff + 1) ? 1.f : 0.f;
      acc = __builtin_amdgcn_wmma_f32_16x16x4_f32(
          /*neg_a=*/false, a, /*neg_b=*/false, B[c],
          /*c_mod=*/(short)0, acc, /*reuse_a=*/false, /*reuse_b=*/false);
    }
    // D layout: VGPR r holds row r (lanes 0-15) / row r+8 (lanes 16-31), col = lane&15
    const int rbase = t * 16 + ((lane < 16) ? 0 : 8);
#pragma unroll
    for (int r = 0; r < 8; ++r)
      sm[(rbase + r) * 17 + col] = acc[r];
  }
  // same-wave LDS RAW: compiler inserts s_wait_dscnt; no cross-wave sharing.

  // ---- gather this centre's 9 selected taps (conflict-free: stride 17) ----
  float V[3][3];
#pragma unroll
  for (int a = 0; a < 3; ++a)
#pragma unroll
    for (int b = 0; b < 3; ++b)
      V[a][b] = sm[lane * 17 + a * 3 + b];

  // ---- per-centre scalars ----
  const float fh = f_centres[m] + zeta_f[m] * INV_PI;
  const float nh = n_centres[m] + zeta_n[m] * INV_PI;
  const float f0 = floorf(fh);
  const float n0 = floorf(nh);
  const float wf = fh - f0;
  const float wn = nh - n0;
  const int f0i = (int)f0;
  const int n0i = (int)n0;
  const float er = expf(log_rho[m]);
  const float th = theta[m];
  const float ar = er * cosf(th);
  const float ai = er * sinf(th);

  // ---- separable merge: 9 taps + shared bilinear frac -> dense 4x4 patch ----
  float T[3][4];
#pragma unroll
  for (int a = 0; a < 3; ++a) {
    T[a][0] = (1.f - wn) * V[a][0];
    T[a][1] = (1.f - wn) * V[a][1] + wn * V[a][0];
    T[a][2] = (1.f - wn) * V[a][2] + wn * V[a][1];
    T[a][3] =                        wn * V[a][2];
  }
  float P4[4][4];
#pragma unroll
  for (int c = 0; c < 4; ++c) {
    P4[0][c] = (1.f - wf) * T[0][c];
    P4[1][c] = (1.f - wf) * T[1][c] + wf * T[0][c];
    P4[2][c] = (1.f - wf) * T[2][c] + wf * T[1][c];
    P4[3][c] =                        wf * T[2][c];
  }

  // ---- scatter: 16 complex cells = 32 f32 atomics (agent scope, L2-resident) ----
#pragma unroll
  for (int r = 0; r < 4; ++r) {
    const int row = f0i - 1 + r;
    if (row < 0 || row >= F_DIM) continue;
#pragma unroll
    for (int c = 0; c < 4; ++c) {
      const int cc = n0i - 1 + c;
      if (cc < 0 || cc >= N_DIM) continue;
      const float w = P4[r][c];
      float* dst = out + 2 * ((size_t)row * N_DIM + cc);
      __hip_atomic_fetch_add(dst,     ar * w, __ATOMIC_RELAXED, __HIP_MEMORY_SCOPE_AGENT);
      __hip_atomic_fetch_add(dst + 1, ai * w, __ATOMIC_RELAXED, __HIP_MEMORY_SCOPE_AGENT);
    }
  }
}

// ---------------------------------------------------------------------------
// Launch
// ---------------------------------------------------------------------------
extern "C" void kernel_launch(void* const* d_in, const int* in_sizes, int n_in,
                              void* d_out, int out_size, void* d_ws, size_t ws_size,
                              hipStream_t stream) {
  (void)in_sizes; (void)n_in; (void)out_size; (void)d_ws; (void)ws_size;

  const float* P_bank    = (const float*)d_in[0];
  const float* alpha     = (const float*)d_in[1];
  const float* zeta_f    = (const float*)d_in[2];
  const float* zeta_n    = (const float*)d_in[3];
  const float* log_rho   = (const float*)d_in[4];
  const float* theta     = (const float*)d_in[5];
  const float* f_centres = (const float*)d_in[6];
  const float* n_centres = (const float*)d_in[7];
  float* out = (float*)d_out;

  // zero the complex grid (2*F*N floats)
  const int n4 = (2 * F_DIM * N_DIM) / 4;
  zero_out_kernel<<<(n4 + 255) / 256, 256, 0, stream>>>((v4f*)out, n4);

  // 256 centres per block (8 waves x 32 centres), fully active waves
  const int blocks = M_DIM / 256;
  codec_splat_kernel<<<blocks, 256, 0, stream>>>(
      P_bank, alpha, zeta_f, zeta_n, log_rho, theta, f_centres, n_centres, out);
}